// ConvAConnect_43782896615960
// MI455X (gfx1250) — compile-verified
//
#include <hip/hip_runtime.h>
#include <hip/hip_bf16.h>

// ---------------------------------------------------------------------------
// ConvAConnect (per-sample noisy-weight 3x3 NHWC conv) for MI455X / gfx1250.
//
// Roofline: 9.7 GFLOP vs ~4.3 GB of HBM traffic (2.25 FLOP/B). At 23.3 TB/s
// only ~52 TFLOP/s of math is needed -> hard memory bound. Therefore:
//   * math: full-precision V_WMMA_F32_16X16X4_F32 (exact fp32, matches the
//     reference numerics; matrix-core throughput is free at this intensity)
//   * input:  GLOBAL_LOAD_ASYNC_TO_LDS_B128 (direct global->LDS, ASYNCcnt,
//     no VGPR round-trip), zero-fill only the SAME-padding halo lanes
//   * output: LDS-transposed so each wave emits ONE full-wave
//     global_store_b128 covering 512 contiguous bytes (full cachelines)
// ---------------------------------------------------------------------------

typedef float v2f __attribute__((ext_vector_type(2)));
typedef float v4f __attribute__((ext_vector_type(4)));
typedef float v8f __attribute__((ext_vector_type(8)));

namespace cfg {
constexpr int Bn   = 128;
constexpr int H    = 256;
constexpr int W    = 256;
constexpr int CIN  = 8;
constexpr int COUT = 8;
constexpr int KTOT = 72;           // 3*3*8
constexpr int TILE_W = 128;        // output pixels per block (8 waves * 16)
constexpr int XCOLS  = TILE_W + 2; // halo columns
}

// Direct global -> LDS async copy (CDNA5 TDM-class path, tracked by ASYNCcnt).
// Inline asm: portable across ROCm 7.2 / amdgpu-toolchain builtin arity skew.
__device__ __forceinline__ void async_load_b128(const float* gp, unsigned lds_byte_off) {
    asm volatile("global_load_async_to_lds_b128 %0, %1, off"
                 :
                 : "v"(lds_byte_off), "v"((unsigned long long)(uintptr_t)gp)
                 : "memory");
}

// Generic pointers to __shared__ are {aperture_hi32, lds_offset32}: the low
// 32 bits are the workgroup-relative LDS byte address the HW wants in VDST.
__device__ __forceinline__ unsigned lds_addr_of(const void* p) {
    return (unsigned)(uintptr_t)p;
}

__global__ __launch_bounds__(256, 2)
void conv_aconnect_wmma(const float* __restrict__ X,
                        const float* __restrict__ Wt,
                        const float* __restrict__ bias,
                        const float* __restrict__ Werr,
                        const float* __restrict__ Berr,
                        float* __restrict__ out)
{
    using namespace cfg;
    __shared__ float xtile[3 * XCOLS * CIN];      // 12480 B: rows h-1..h+1 halo tile
    __shared__ float effW[KTOT * 16];             // 4608 B: W*Werr[b], cols 8..15 = 0
    __shared__ float effB[COUT];
    __shared__ float ostage[8][16 * COUT];        // 4096 B: per-wave store staging

    const int tid   = threadIdx.x;
    const int wtile = blockIdx.x;      // 0..W/TILE_W-1
    const int h     = blockIdx.y;      // 0..H-1
    const int b     = blockIdx.z;      // 0..B-1
    const int wbase = wtile * TILE_W;

    // ---- Stage X halo tile: async global->LDS b128; zero-store OOB halo ----
    constexpr int NV4 = 3 * XCOLS * (CIN / 4);    // 780 float4 chunks
    for (int i = tid; i < NV4; i += 256) {
        int r    = i / (XCOLS * 2);
        int rem  = i - r * (XCOLS * 2);
        int c    = rem >> 1;
        int half = rem & 1;
        int hin  = h + r - 1;
        int win  = wbase + c - 1;
        float* lp = &xtile[(r * XCOLS + c) * CIN + half * 4];
        if ((unsigned)hin < (unsigned)H && (unsigned)win < (unsigned)W) {
            const float* gp = X + (((size_t)b * H + hin) * W + win) * CIN + half * 4;
            async_load_b128(gp, lds_addr_of(lp));
        } else {
            v4f z = {0.f, 0.f, 0.f, 0.f};
            *(v4f*)lp = z;                        // SAME-padding zeros
        }
    }

    // ---- Per-sample effective weights, padded to 16 N-columns (zeros) -----
    for (int i = tid; i < KTOT * 16; i += 256) {
        int k = i >> 4;
        int n = i & 15;
        float v = 0.f;
        if (n < COUT) {
            int wi = k * COUT + n;
            v = Wt[wi] * Werr[(size_t)b * (KTOT * COUT) + wi];
        }
        effW[i] = v;
    }
    if (tid < COUT) effB[tid] = bias[tid] * Berr[b * COUT + tid];

    asm volatile("s_wait_asynccnt 0x0" ::: "memory");   // async tile resident in LDS
    __syncthreads();

    // ---- WMMA: 16 pixels (M) x 16 (N; 8 valid) tile per wave, K = 72 ------
    const int wave   = tid >> 5;
    const int lane   = tid & 31;
    const int m      = lane & 15;              // A: M row / B,C: N column
    const int khalf  = (lane >> 4) << 1;       // 0 (lanes 0-15) or 2 (16-31)
    const int wlocal = wave * 16 + m;          // pixel column inside LDS tile

    v8f acc = {0.f, 0.f, 0.f, 0.f, 0.f, 0.f, 0.f, 0.f};

    #pragma unroll
    for (int s = 0; s < KTOT / 4; ++s) {       // 18 chained v_wmma_f32_16x16x4_f32
        const int k0 = s * 4;
        v2f a, bf;
        #pragma unroll
        for (int v = 0; v < 2; ++v) {
            int k  = k0 + khalf + v;           // this lane's K index (ISA A layout)
            int ci = k & 7;
            int t  = k >> 3;                   // 0..8 = kh*3+kw
            int kh = (t * 11) >> 5;            // t / 3 for t in [0,8]
            int kw = t - 3 * kh;
            a[v]  = xtile[(kh * XCOLS + wlocal + kw) * CIN + ci];  // im2col from LDS
            bf[v] = effW[k * 16 + m];                              // B row striped on lanes
        }
        // 8 args: (neg_a, A, neg_b, B, c_mod, C, reuse_a, reuse_b)
        acc = __builtin_amdgcn_wmma_f32_16x16x4_f32(
            false, a, false, bf, (short)0, acc, false, false);
    }

    // ---- Store: transpose through per-wave LDS stage, then one full-wave
    //      global_store_b128 of 512 contiguous bytes (NHWC is pixel-major) --
    const int n = m;
    if (n < COUT) {
        const int mbase = (lane >> 4) * 8;     // hi lane-half holds M = r + 8
        const float bv = effB[n];
        #pragma unroll
        for (int r = 0; r < 8; ++r) {
            ostage[wave][(mbase + r) * COUT + n] = acc[r] + bv;    // 16 lanes cover all 128
        }
    }
    __builtin_amdgcn_wave_barrier();           // keep LDS store->load order (same wave,
                                               // DS is in-order; this blocks reordering)
    v4f ov = *(const v4f*)&ostage[wave][lane * 4];

    const size_t outbase = (((size_t)b * H + h) * W + wbase) * COUT;
    *(v4f*)(out + outbase + (size_t)wave * (16 * COUT) + lane * 4) = ov;
}

extern "C" void kernel_launch(void* const* d_in, const int* in_sizes, int n_in,
                              void* d_out, int out_size, void* d_ws, size_t ws_size,
                              hipStream_t stream) {
    using namespace cfg;
    const float* X    = (const float*)d_in[0];
    const float* Wt   = (const float*)d_in[1];
    const float* bias = (const float*)d_in[2];
    const float* Werr = (const float*)d_in[3];
    const float* Berr = (const float*)d_in[4];
    float* out = (float*)d_out;

    dim3 grid(W / TILE_W, H, Bn);   // 2 x 256 x 128 = 65536 blocks
    conv_aconnect_wmma<<<grid, 256, 0, stream>>>(X, Wt, bias, Werr, Berr, out);
}